// HGTEmbeddingModule_30923764532054
// MI455X (gfx1250) — compile-verified
//
#include <hip/hip_runtime.h>
#include <hip/hip_bf16.h>
#include <math.h>

// ---------------------------------------------------------------------------
// HGT forward for MI455X (gfx1250, wave32, WMMA).
// All dense GEMMs (N=128) run through V_WMMA_F32_16X16X4_F32.
// Per-relation [H,D,D] einsums are folded into the K/V projection weights.
// Edge segment-softmax uses ordered-uint atomicMax + f32 atomicAdd.
// ---------------------------------------------------------------------------

typedef __attribute__((ext_vector_type(2))) float v2f;
typedef __attribute__((ext_vector_type(8))) float v8f;

#define HID 128
#define NHEAD 4
#define HDIM 32

__device__ __forceinline__ float gelu_exact(float x) {
    return 0.5f * x * (1.0f + erff(x * 0.70710678118654752440f));
}

// order-preserving encoding of float into uint32 for atomicMax
__device__ __forceinline__ unsigned ord_encode(float f) {
    unsigned u = __float_as_uint(f);
    return (u & 0x80000000u) ? ~u : (u | 0x80000000u);
}
__device__ __forceinline__ float ord_decode(unsigned u) {
    u = (u & 0x80000000u) ? (u & 0x7FFFFFFFu) : ~u;
    return __uint_as_float(u);
}

// ---------------------------------------------------------------------------
// GEMM: out[M,128] = epilogue( preact(A[M,K]) @ W[K,128] + bias )
// MODE bit0: post-ReLU
// MODE bit1: pre-GELU on A (after scaling by pre_scale)  [mean+gelu fusion]
// MODE bit2: skip-mix: out = sigmoid(*skip_p)*val + (1-s)*skip_x
// Block: 128 threads = 4 waves; each wave computes a 16x128 strip.
// W is staged in LDS pair-interleaved so each B fragment is one ds_load_b64.
// WMMA f32 16x16x4 fragment layout (ISA 7.12.2):
//   A: lane<16 -> M=lane, regs {K0,K1}; lane>=16 -> M=lane-16, regs {K2,K3}
//   B: lane<16 -> N=lane, regs {K0,K1}; lane>=16 -> N=lane-16, regs {K2,K3}
//   C/D: reg r: lane<16 -> (M=r, N=lane); lane>=16 -> (M=r+8, N=lane-16)
// MODE is a template parameter so non-GELU GEMMs get a branch-free WMMA loop.
// ---------------------------------------------------------------------------
template <int MODE>
__global__ __launch_bounds__(128)
void gemm_hid_kernel(const float* __restrict__ A, int M, int K,
                     const float* __restrict__ W, const float* __restrict__ bias,
                     const float* __restrict__ skip_x, const float* __restrict__ skip_p,
                     float* __restrict__ out, float pre_scale)
{
    __shared__ float ldsW[128 * HID];   // 64 KB: [K/2][128][2] pair-interleaved
    const int tid = threadIdx.x;

    // Stage W: global row-major [K][128] -> LDS pair-interleaved.
    const int nW4 = (K * HID) >> 2;
    const float4* Wv = (const float4*)W;
    for (int i = tid; i < nW4; i += 128) {
        float4 w = Wv[i];
        int k = (i * 4) / HID;
        int c = (i * 4) % HID;
        int p = k >> 1, lo = k & 1;
        ldsW[p * 256 + (c + 0) * 2 + lo] = w.x;
        ldsW[p * 256 + (c + 1) * 2 + lo] = w.y;
        ldsW[p * 256 + (c + 2) * 2 + lo] = w.z;
        ldsW[p * 256 + (c + 3) * 2 + lo] = w.w;
    }
    __syncthreads();

    const int wave = tid >> 5;
    const int lane = tid & 31;
    const int half = lane >> 4;     // 0 or 1
    const int l16  = lane & 15;

    int rowBase = blockIdx.x * 64 + wave * 16;
    int row = rowBase + l16;
    int rowc = (row < M) ? row : (M - 1);     // clamp loads; stores are guarded
    const float* arow = A + (size_t)rowc * K;

    // prefetch this lane's A row (global_prefetch_b8; K*4 bytes spans <=4 lines)
    __builtin_prefetch(arow, 0, 3);
    __builtin_prefetch(arow + 32, 0, 3);
    __builtin_prefetch(arow + 64, 0, 3);
    __builtin_prefetch(arow + 96, 0, 3);

    v8f acc[8];
    #pragma unroll
    for (int n = 0; n < 8; ++n) {
        #pragma unroll
        for (int r = 0; r < 8; ++r) acc[n][r] = 0.0f;
    }

    const v2f* lds2 = (const v2f*)ldsW;
    const int ksteps = K >> 2;                 // 16 or 32, always even
    for (int ks = 0; ks < ksteps; ks += 2) {
        int kk = ks * 4 + half * 2;
        v2f a0, a1;
        a0[0] = arow[kk];
        a0[1] = arow[kk + 1];
        a1[0] = arow[kk + 4];
        a1[1] = arow[kk + 5];
        if (MODE & 2) {
            a0[0] = gelu_exact(a0[0] * pre_scale);
            a0[1] = gelu_exact(a0[1] * pre_scale);
            a1[0] = gelu_exact(a1[0] * pre_scale);
            a1[1] = gelu_exact(a1[1] * pre_scale);
        }
        int p0 = ks * 2 + half;
        #pragma unroll
        for (int n = 0; n < 8; ++n) {
            v2f b = lds2[p0 * 128 + n * 16 + l16];
            acc[n] = __builtin_amdgcn_wmma_f32_16x16x4_f32(
                false, a0, false, b, (short)0, acc[n], false, false);
        }
        #pragma unroll
        for (int n = 0; n < 8; ++n) {
            v2f b = lds2[(p0 + 2) * 128 + n * 16 + l16];
            acc[n] = __builtin_amdgcn_wmma_f32_16x16x4_f32(
                false, a1, false, b, (short)0, acc[n], false, false);
        }
    }

    float gate = 0.0f, one_m = 1.0f;
    if (MODE & 4) {
        float sp = *skip_p;
        gate = 1.0f / (1.0f + __expf(-sp));
        one_m = 1.0f - gate;
    }

    #pragma unroll
    for (int n = 0; n < 8; ++n) {
        int col = n * 16 + l16;
        float bv = bias ? bias[col] : 0.0f;
        #pragma unroll
        for (int r = 0; r < 8; ++r) {
            int orow = rowBase + r + half * 8;
            if (orow < M) {
                float v = acc[n][r] + bv;
                if (MODE & 1) v = fmaxf(v, 0.0f);
                if (MODE & 4) v = gate * v + one_m * skip_x[(size_t)orow * HID + col];
                out[(size_t)orow * HID + col] = v;
            }
        }
    }
}

// ---------------------------------------------------------------------------
// Fold per-relation transform into projection weights:
//   Wc[i][h*32+e] = sum_d Wsrc[i][h*32+d] * rel[h][d][e]   (i<K)
//   bc[h*32+e]    = sum_d bsrc[h*32+d]    * rel[h][d][e]   (block i==K)
// ---------------------------------------------------------------------------
__global__ __launch_bounds__(128)
void fold_rel_kernel(const float* __restrict__ Wsrc, const float* __restrict__ bsrc,
                     const float* __restrict__ rel,
                     float* __restrict__ Wc, float* __restrict__ bc, int K)
{
    int j = threadIdx.x;                 // output column 0..127
    int i = blockIdx.x;                  // row, or K for bias
    int h = j >> 5, e = j & 31;
    const float* r = rel + h * (HDIM * HDIM) + e;          // stride 32 over d
    const float* src = ((i < K) ? (Wsrc + (size_t)i * HID) : bsrc) + h * HDIM;
    float s = 0.0f;
    #pragma unroll
    for (int d = 0; d < HDIM; ++d) s += src[d] * r[d * HDIM];
    if (i < K) Wc[(size_t)i * HID + j] = s;
    else       bc[j] = s;
}

// ---------------------------------------------------------------------------
__global__ void fill_u32_kernel(unsigned* __restrict__ p, int n, unsigned val) {
    int i = blockIdx.x * blockDim.x + threadIdx.x;
    if (i < n) p[i] = val;
}

// ---------------------------------------------------------------------------
// Pass 1: per-(edge,head) attention logit + atomic segment max.
// ---------------------------------------------------------------------------
__global__ __launch_bounds__(256)
void edge_alpha_kernel(const int* __restrict__ ei, int E,
                       const float* __restrict__ q, const float* __restrict__ krel,
                       const float* __restrict__ p_rel, float scale,
                       float* __restrict__ alpha, unsigned* __restrict__ amax)
{
    int gt = blockIdx.x * 256 + threadIdx.x;
    int e = gt >> 2, h = gt & 3;
    if (e >= E) return;
    int si = ei[e];
    int di = ei[E + e];
    const float4* qv = (const float4*)(q    + (size_t)di * HID + h * HDIM);
    const float4* kv = (const float4*)(krel + (size_t)si * HID + h * HDIM);
    float s = 0.0f;
    #pragma unroll
    for (int i = 0; i < 8; ++i) {
        float4 a = qv[i], b = kv[i];
        s += a.x * b.x + a.y * b.y + a.z * b.z + a.w * b.w;
    }
    s *= p_rel[h] * scale;
    alpha[(size_t)e * 4 + h] = s;
    atomicMax(&amax[(size_t)di * 4 + h], ord_encode(s));
}

// ---------------------------------------------------------------------------
// Pass 2: ea = exp(alpha - max); atomic segment sum. alpha overwritten by ea.
// ---------------------------------------------------------------------------
__global__ __launch_bounds__(256)
void edge_exp_kernel(const int* __restrict__ ei, int E,
                     const unsigned* __restrict__ amax,
                     float* __restrict__ alpha, float* __restrict__ denom)
{
    int gt = blockIdx.x * 256 + threadIdx.x;
    int e = gt >> 2, h = gt & 3;
    if (e >= E) return;
    int di = ei[E + e];
    float m = ord_decode(amax[(size_t)di * 4 + h]);
    float ea = __expf(alpha[(size_t)e * 4 + h] - m);
    alpha[(size_t)e * 4 + h] = ea;
    atomicAdd(&denom[(size_t)di * 4 + h], ea);
}

// ---------------------------------------------------------------------------
// Pass 3: weighted message scatter-add. One wave32 per edge; each lane does
// one element per head -> 128 consecutive atomic f32 adds per edge.
// ---------------------------------------------------------------------------
__global__ __launch_bounds__(256)
void edge_msg_kernel(const int* __restrict__ ei, int E,
                     const float* __restrict__ alpha, const float* __restrict__ denom,
                     const float* __restrict__ vrel, float* __restrict__ acc)
{
    int lane = threadIdx.x & 31;
    int e = blockIdx.x * 8 + (threadIdx.x >> 5);
    if (e >= E) return;
    int si = ei[e];
    int di = ei[E + e];
    const float* v = vrel + (size_t)si * HID;
    float* a = acc + (size_t)di * HID;
    #pragma unroll
    for (int h = 0; h < NHEAD; ++h) {
        float w = alpha[(size_t)e * 4 + h] / (denom[(size_t)di * 4 + h] + 1e-16f);
        atomicAdd(&a[h * HDIM + lane], w * v[h * HDIM + lane]);
    }
}

// ---------------------------------------------------------------------------
// Host orchestration
// ---------------------------------------------------------------------------
static inline void launch_gemm(const float* A, int M, int K, const float* W,
                               const float* bias, const float* skip_x,
                               const float* skip_p, float* out, int mode,
                               float pre_scale, hipStream_t stream)
{
    dim3 grid((M + 63) / 64), block(128);
    switch (mode) {
    case 0:
        gemm_hid_kernel<0><<<grid, block, 0, stream>>>(A, M, K, W, bias, skip_x,
                                                       skip_p, out, pre_scale);
        break;
    case 1:
        gemm_hid_kernel<1><<<grid, block, 0, stream>>>(A, M, K, W, bias, skip_x,
                                                       skip_p, out, pre_scale);
        break;
    default:  // 6 = pre-GELU | skip-mix
        gemm_hid_kernel<6><<<grid, block, 0, stream>>>(A, M, K, W, bias, skip_x,
                                                       skip_p, out, pre_scale);
        break;
    }
}

static inline void launch_fill_u32(void* p, size_t n, unsigned val, hipStream_t stream) {
    dim3 grid((unsigned)((n + 255) / 256)), block(256);
    fill_u32_kernel<<<grid, block, 0, stream>>>((unsigned*)p, (int)n, val);
}

extern "C" void kernel_launch(void* const* d_in, const int* in_sizes, int n_in,
                              void* d_out, int out_size, void* d_ws, size_t ws_size,
                              hipStream_t stream)
{
    (void)n_in; (void)ws_size; (void)out_size;

    const int NP = in_sizes[0] / 128;     // 100000
    const int NA = in_sizes[1] / 64;      // 50000
    const int E_PP = in_sizes[2] / 2;     // 800000
    const int E_AP = in_sizes[3] / 2;     // 400000
    const int E_PA = in_sizes[4] / 2;     // 400000
    const int BS_PAPER = 8192, BS_AUTHOR = 4096;

    const float* x_paper  = (const float*)d_in[0];
    const float* x_author = (const float*)d_in[1];
    const int*   ei_pp    = (const int*)d_in[2];
    const int*   ei_ap    = (const int*)d_in[3];
    const int*   ei_pa    = (const int*)d_in[4];
    auto P = [&](int i) { return (const float*)d_in[i]; };

    // JAX pytree leaf order (sorted dict keys):
    //   params.layers[l]: et{cites(a,m,p), written_by(a,m,p), writes(a,m,p)},
    //                     nt{author(a.b,a.w,k.b,k.w,q.b,q.w,skip,v.b,v.w),
    //                        paper (same)}                       -> 27 leaves
    //   params.lin: author.b, author.w, paper.b, paper.w
    const int LIN_A_B = 5 + 54 + 0, LIN_A_W = 5 + 54 + 1;
    const int LIN_P_B = 5 + 54 + 2, LIN_P_W = 5 + 54 + 3;

    // workspace carve-up (fp32 elements)
    float* ws = (float*)d_ws;
    size_t off = 0;
    auto alloc = [&](size_t n) { float* p = ws + off; off += n; return p; };
    float* xp[2] = { alloc((size_t)NP * HID), alloc((size_t)NP * HID) };
    float* xa[2] = { alloc((size_t)NA * HID), alloc((size_t)NA * HID) };
    float* qp   = alloc((size_t)NP * HID);
    float* qa   = alloc((size_t)NA * HID);
    float* krel = alloc((size_t)NP * HID);      // sized for max src count
    float* vrel = alloc((size_t)NP * HID);
    float* accp = alloc((size_t)NP * HID);
    float* acca = alloc((size_t)NA * HID);
    float* alpha = alloc((size_t)E_PP * NHEAD); // sized for max E
    unsigned* amax = (unsigned*)alloc((size_t)NP * NHEAD);
    float* denom = alloc((size_t)NP * NHEAD);
    float* Wck = alloc(128 * HID); float* bck = alloc(HID);
    float* Wcv = alloc(128 * HID); float* bcv = alloc(HID);

    const float scale = 0.17677669529663687f;   // 1/sqrt(32)
    const unsigned NEG_INF_ORD = 0x007FFFFFu;   // ord_encode(-inf)
    const unsigned ZERO_F = 0u;                  // bits of 0.0f

    // input projections: x = relu(X @ W + b)
    launch_gemm(x_paper,  NP, 128, P(LIN_P_W), P(LIN_P_B), nullptr, nullptr,
                xp[0], /*relu*/1, 1.0f, stream);
    launch_gemm(x_author, NA,  64, P(LIN_A_W), P(LIN_A_B), nullptr, nullptr,
                xa[0], 1, 1.0f, stream);

    int cur = 0;
    for (int l = 0; l < 2; ++l) {
        const int L = 5 + 27 * l;
        // et leaves: cites L+0..2, written_by L+3..5, writes L+6..8
        const float* a_rel[3] = { P(L + 0), P(L + 3), P(L + 6) };
        const float* m_rel[3] = { P(L + 1), P(L + 4), P(L + 7) };
        const float* p_rel[3] = { P(L + 2), P(L + 5), P(L + 8) };
        // nt.author: a.b L+9, a.w L+10, k.b L+11, k.w L+12, q.b L+13, q.w L+14,
        //            skip L+15, v.b L+16, v.w L+17 ; nt.paper: +18..26
        const float* AkB = P(L + 11); const float* AkW = P(L + 12);
        const float* AqB = P(L + 13); const float* AqW = P(L + 14);
        const float* AvB = P(L + 16); const float* AvW = P(L + 17);
        const float* PkB = P(L + 20); const float* PkW = P(L + 21);
        const float* PqB = P(L + 22); const float* PqW = P(L + 23);
        const float* PvB = P(L + 25); const float* PvW = P(L + 26);

        // Q projections (dst side, plain)
        launch_gemm(xp[cur], NP, 128, PqW, PqB, nullptr, nullptr, qp, 0, 1.0f, stream);
        launch_gemm(xa[cur], NA, 128, AqW, AqB, nullptr, nullptr, qa, 0, 1.0f, stream);

        // zero accumulators
        launch_fill_u32(accp, (size_t)NP * HID, ZERO_F, stream);
        launch_fill_u32(acca, (size_t)NA * HID, ZERO_F, stream);

        // edge type table: {ei, E, src(x,W,b,N), dst(q,acc,N)}
        const int*   eis[3]   = { ei_pp, ei_pa, ei_ap };
        const int    Es[3]    = { E_PP, E_PA, E_AP };
        const float* xsrc[3]  = { xp[cur], xp[cur], xa[cur] };
        const int    Nsrc[3]  = { NP, NP, NA };
        const float* WkS[3]   = { PkW, PkW, AkW };
        const float* bkS[3]   = { PkB, PkB, AkB };
        const float* WvS[3]   = { PvW, PvW, AvW };
        const float* bvS[3]   = { PvB, PvB, AvB };
        const float* qdst[3]  = { qp, qa, qp };
        float*       accd[3]  = { accp, acca, accp };
        const int    Ndst[3]  = { NP, NA, NP };

        for (int et = 0; et < 3; ++et) {
            const int E = Es[et];
            // fold relation matrices into K/V projection weights
            fold_rel_kernel<<<dim3(129), dim3(128), 0, stream>>>(
                WkS[et], bkS[et], a_rel[et], Wck, bck, 128);
            fold_rel_kernel<<<dim3(129), dim3(128), 0, stream>>>(
                WvS[et], bvS[et], m_rel[et], Wcv, bcv, 128);
            // krel/vrel over source nodes (WMMA GEMMs)
            launch_gemm(xsrc[et], Nsrc[et], 128, Wck, bck, nullptr, nullptr,
                        krel, 0, 1.0f, stream);
            launch_gemm(xsrc[et], Nsrc[et], 128, Wcv, bcv, nullptr, nullptr,
                        vrel, 0, 1.0f, stream);
            // segment softmax state
            launch_fill_u32(amax, (size_t)Ndst[et] * NHEAD, NEG_INF_ORD, stream);
            launch_fill_u32(denom, (size_t)Ndst[et] * NHEAD, ZERO_F, stream);
            // pass 1: logits + max
            {
                dim3 grid((unsigned)(((size_t)E * 4 + 255) / 256)), block(256);
                edge_alpha_kernel<<<grid, block, 0, stream>>>(
                    eis[et], E, qdst[et], krel, p_rel[et], scale, alpha, amax);
            }
            // pass 2: exp + denom
            {
                dim3 grid((unsigned)(((size_t)E * 4 + 255) / 256)), block(256);
                edge_exp_kernel<<<grid, block, 0, stream>>>(
                    eis[et], E, amax, alpha, denom);
            }
            // pass 3: weighted message scatter
            {
                dim3 grid((unsigned)((E + 7) / 8)), block(256);
                edge_msg_kernel<<<grid, block, 0, stream>>>(
                    eis[et], E, alpha, denom, vrel, accd[et]);
            }
        }

        // out = gelu(mean(acc)) @ Wa + ba ; x' = s*out + (1-s)*x
        // paper receives 2 edge types -> mean scale 0.5; author 1 -> 1.0
        launch_gemm(accp, NP, 128, P(L + 19), P(L + 18), xp[cur], P(L + 24),
                    xp[1 - cur], /*gelu|skip*/ 2 | 4, 0.5f, stream);
        launch_gemm(acca, NA, 128, P(L + 10), P(L + 9), xa[cur], P(L + 15),
                    xa[1 - cur], 2 | 4, 1.0f, stream);
        cur ^= 1;
    }

    // outputs: x_paper[:8192] then x_author[:4096], concatenated
    float* outp = (float*)d_out;
    hipMemcpyAsync(outp, xp[cur], (size_t)BS_PAPER * HID * sizeof(float),
                   hipMemcpyDeviceToDevice, stream);
    hipMemcpyAsync(outp + (size_t)BS_PAPER * HID, xa[cur],
                   (size_t)BS_AUTHOR * HID * sizeof(float),
                   hipMemcpyDeviceToDevice, stream);
}